// DeepFM_77558519431939
// MI455X (gfx1250) — compile-verified
//
#include <hip/hip_runtime.h>
#include <math.h>

typedef __attribute__((ext_vector_type(2))) float v2f;
typedef __attribute__((ext_vector_type(8))) float v8f;

#define B_ROWS   16384
#define EMB      10
#define N_DENSE  7
#define N_FIELDS 8
#define DEEP_IN  90
#define KPAD     96      // 90 padded to multiple of 4 (and of 8)
#define HID      128
#define S_TOTAL  1866000

__constant__ int c_offsets[N_FIELDS] = {
    0, 1000000, 1500000, 1700000, 1800000, 1850000, 1860000, 1865000
};

// ---------------------------------------------------------------------------
// Kernel 1: per-row fused embedding gather, dense embedding, first-order
// linear term, and FM cross term.  Writes the zero-padded [B, 96] deep input
// and the scalar (lin + cross) partial per row.
// ---------------------------------------------------------------------------
__global__ __launch_bounds__(256)
void deepfm_row(const int*   __restrict__ sparse,     // [B, 8]
                const float* __restrict__ dense,      // [B, 7]
                const float* __restrict__ emb_table,  // [S, 10]
                const float* __restrict__ W_dense,    // [10, 7]
                const float* __restrict__ b_dense,    // [10]
                const float* __restrict__ w_first,    // [S + 7]
                const float* __restrict__ b_first,    // [1]
                float*       __restrict__ flatA,      // [B, 96]
                float*       __restrict__ partial)    // [B]
{
    const int row = blockIdx.x * blockDim.x + threadIdx.x;
    if (row >= B_ROWS) return;

    float d[N_DENSE];
#pragma unroll
    for (int i = 0; i < N_DENSE; ++i) d[i] = dense[row * N_DENSE + i];

    // dense_emb = dense @ W_dense.T + b_dense   (W_dense is [10,7] row-major)
    float de[EMB];
#pragma unroll
    for (int j = 0; j < EMB; ++j) {
        float acc = b_dense[j];
#pragma unroll
        for (int i = 0; i < N_DENSE; ++i)
            acc = fmaf(d[i], W_dense[j * N_DENSE + i], acc);
        de[j] = acc;
    }

    float s[EMB];
#pragma unroll
    for (int j = 0; j < EMB; ++j) s[j] = 0.0f;
    float ss  = 0.0f;
    float lin = 0.0f;

    float* frow = flatA + (size_t)row * KPAD;

#pragma unroll
    for (int f = 0; f < N_FIELDS; ++f) {
        const int gid = sparse[row * N_FIELDS + f] + c_offsets[f];
        const float* e = emb_table + (size_t)gid * EMB;
        lin += w_first[gid];
#pragma unroll
        for (int j = 0; j < EMB; ++j) {
            const float v = e[j];
            s[j] += v;
            ss   = fmaf(v, v, ss);
            frow[f * EMB + j] = v;
        }
    }

    // append dense embedding as field 8
#pragma unroll
    for (int j = 0; j < EMB; ++j) {
        const float v = de[j];
        s[j] += v;
        ss   = fmaf(v, v, ss);
        frow[N_FIELDS * EMB + j] = v;
    }
    // zero-pad K = 90..95 so the WMMA K-loop needs no masking on A
#pragma unroll
    for (int k = DEEP_IN; k < KPAD; ++k) frow[k] = 0.0f;

    // first-order dense part
#pragma unroll
    for (int i = 0; i < N_DENSE; ++i)
        lin = fmaf(d[i], w_first[S_TOTAL + i], lin);
    lin += b_first[0];

    float s2 = 0.0f;
#pragma unroll
    for (int j = 0; j < EMB; ++j) s2 = fmaf(s[j], s[j], s2);
    const float cross = 0.5f * (s2 - ss);

    partial[row] = lin + cross;
}

// ---------------------------------------------------------------------------
// Kernels 2/3: fp32 WMMA GEMM + bias + ReLU, LDS-staged, branch-free K-loop.
//   out[B, 128] = relu(A[B, KDIM] (first KVALID cols) @ W[KVALID, 128] + bias)
//
// Block = 256 threads = 8 waves; wave w owns cols [16w, 16w+16); block covers
// 16 rows per tile, persistent over tiles (grid-stride).  W is staged once per
// block into LDS with rows KVALID..KDIM-1 zeroed, so the fully unrolled
// V_WMMA_F32_16X16X4_F32 K-loop needs no masks (EXEC stays all-ones).
//
// Fragment layouts per CDNA5 ISA 7.12.2 (32-bit, wave32):
//   A 16x4 : lanes 0-15 M=0..15 {K+0,K+1}; lanes 16-31 M=0..15 {K+2,K+3}
//   B 4x16 : lanes 0-15 N=0..15 {K+0,K+1}; lanes 16-31 N=0..15 {K+2,K+3}
//   C/D    : VGPR r -> M=r (lanes 0-15) / M=8+r (lanes 16-31), N=lane&15
//
// LDS strides: A row stride KDIM+2 (even, kills the 32-bank stride conflict);
// W row stride 136 so rows k and k+2 hit disjoint bank groups (2*136%64==16).
// ---------------------------------------------------------------------------
template <int KDIM, int KVALID>
__global__ __launch_bounds__(256)
void gemm_relu_wmma(const float* __restrict__ A,     // [B, KDIM]
                    const float* __restrict__ W,     // [KVALID, 128] row-major
                    const float* __restrict__ bias,  // [128]
                    float*       __restrict__ out,   // [B, 128]
                    int ntiles)
{
    constexpr int ASTR = KDIM + 2;
    constexpr int WSTR = 136;
    __shared__ float Wlds[KDIM * WSTR];
    __shared__ float Alds[16 * ASTR];

    const int tid  = threadIdx.x;
    const int lane = tid & 31;
    const int wave = tid >> 5;
    const int m    = lane & 15;
    const int koff = (lane >> 4) * 2;
    const int col  = wave * 16 + m;

    // ---- stage W into LDS (zero-pad rows KVALID..KDIM-1), float4 chunks ----
    constexpr int WCHUNKS = (KDIM * 128) / 4;
    for (int c = tid; c < WCHUNKS; c += 256) {
        const int el = c * 4;
        const int k  = el >> 7;       // row
        const int cc = el & 127;      // col (multiple of 4, stays in-row)
        float4 v = make_float4(0.f, 0.f, 0.f, 0.f);
        if (el < KVALID * 128) v = *(const float4*)(W + el);
        Wlds[k * WSTR + cc + 0] = v.x;
        Wlds[k * WSTR + cc + 1] = v.y;
        Wlds[k * WSTR + cc + 2] = v.z;
        Wlds[k * WSTR + cc + 3] = v.w;
    }

    const float bcol  = bias[col];
    const int   rbase = (lane >> 4) * 8;

    for (int tile = blockIdx.x; tile < ntiles; tile += gridDim.x) {
        const int    row0  = tile * 16;
        const float* Atile = A + (size_t)row0 * KDIM;   // contiguous 16*KDIM

        __syncthreads();   // Wlds ready (iter 0) / previous tile's reads done

        // prefetch next A tile toward L2 while we fill + compute this one
        if (tile + (int)gridDim.x < ntiles) {
            const float* Anext = A + (size_t)(tile + gridDim.x) * 16 * KDIM;
            __builtin_prefetch(Anext + tid * (16 * KDIM / 256), 0, 0);
        }

        // ---- stage A tile into LDS ----
        constexpr int ACHUNKS = (16 * KDIM) / 4;
        for (int c = tid; c < ACHUNKS; c += 256) {
            const int el = c * 4;
            const int r  = el / KDIM;
            const int k  = el - r * KDIM;   // multiple of 4, stays in-row
            const float4 v = *(const float4*)(Atile + el);
            Alds[r * ASTR + k + 0] = v.x;
            Alds[r * ASTR + k + 1] = v.y;
            Alds[r * ASTR + k + 2] = v.z;
            Alds[r * ASTR + k + 3] = v.w;
        }
        __syncthreads();

        // ---- branch-free WMMA K-loop, fully unrolled ----
        v8f acc = {};
#pragma unroll
        for (int kk = 0; kk < KDIM; kk += 4) {
            const int k0 = kk + koff;
            v2f a, b;
            a.x = Alds[m * ASTR + k0];
            a.y = Alds[m * ASTR + k0 + 1];
            b.x = Wlds[(k0)     * WSTR + col];
            b.y = Wlds[(k0 + 1) * WSTR + col];
            // (neg_a, A, neg_b, B, c_mod, C, reuse_a, reuse_b)
            acc = __builtin_amdgcn_wmma_f32_16x16x4_f32(
                      false, a, false, b, (short)0, acc, false, false);
        }

        // ---- bias + ReLU + store (D-matrix striping) ----
#pragma unroll
        for (int r = 0; r < 8; ++r) {
            float v = acc[r] + bcol;
            v = v > 0.0f ? v : 0.0f;
            out[(size_t)(row0 + rbase + r) * HID + col] = v;
        }
    }
}

// ---------------------------------------------------------------------------
// Kernel 4: deep = h2 @ W3 + b3 ; out = sigmoid(partial + deep)
// ---------------------------------------------------------------------------
__global__ __launch_bounds__(256)
void deepfm_final(const float* __restrict__ h2,      // [B, 128]
                  const float* __restrict__ W3,      // [128]
                  const float* __restrict__ b3,      // [1]
                  const float* __restrict__ partial, // [B]
                  float*       __restrict__ out)     // [B]
{
    const int row = blockIdx.x * blockDim.x + threadIdx.x;
    if (row >= B_ROWS) return;

    const float4* h = (const float4*)(h2 + (size_t)row * HID);
    const float4* w = (const float4*)W3;
    float acc = 0.0f;
#pragma unroll
    for (int k = 0; k < HID / 4; ++k) {
        const float4 hv = h[k];
        const float4 wv = w[k];
        acc = fmaf(hv.x, wv.x, acc);
        acc = fmaf(hv.y, wv.y, acc);
        acc = fmaf(hv.z, wv.z, acc);
        acc = fmaf(hv.w, wv.w, acc);
    }
    const float z = acc + b3[0] + partial[row];
    out[row] = 1.0f / (1.0f + __expf(-z));
}

// ---------------------------------------------------------------------------
extern "C" void kernel_launch(void* const* d_in, const int* in_sizes, int n_in,
                              void* d_out, int out_size, void* d_ws, size_t ws_size,
                              hipStream_t stream)
{
    const int*   sparse    = (const int*)  d_in[0];   // [B, 8]
    const float* dense     = (const float*)d_in[1];   // [B, 7]
    const float* emb_table = (const float*)d_in[2];   // [S, 10]
    const float* W_dense   = (const float*)d_in[3];   // [10, 7]
    const float* b_dense   = (const float*)d_in[4];   // [10]
    const float* w_first   = (const float*)d_in[5];   // [S + 7]
    const float* b_first   = (const float*)d_in[6];   // [1]
    const float* W1        = (const float*)d_in[7];   // [90, 128]
    const float* b1        = (const float*)d_in[8];   // [128]
    const float* W2        = (const float*)d_in[9];   // [128, 128]
    const float* b2        = (const float*)d_in[10];  // [128]
    const float* W3        = (const float*)d_in[11];  // [128, 1]
    const float* b3        = (const float*)d_in[12];  // [1]
    float*       out       = (float*)d_out;

    // workspace carve-up
    char* ws = (char*)d_ws;
    float* flatA   = (float*)ws;  ws += (size_t)B_ROWS * KPAD * sizeof(float);
    float* h1      = (float*)ws;  ws += (size_t)B_ROWS * HID  * sizeof(float);
    float* h2      = (float*)ws;  ws += (size_t)B_ROWS * HID  * sizeof(float);
    float* partial = (float*)ws;

    const int ntiles = B_ROWS / 16;   // 1024

    // 1) fused per-row gather / dense-emb / first-order / FM cross
    deepfm_row<<<B_ROWS / 256, 256, 0, stream>>>(
        sparse, dense, emb_table, W_dense, b_dense, w_first, b_first,
        flatA, partial);

    // 2) h1 = relu(flat @ W1 + b1)   (K padded 90 -> 96)
    gemm_relu_wmma<KPAD, DEEP_IN><<<256, 256, 0, stream>>>(
        flatA, W1, b1, h1, ntiles);

    // 3) h2 = relu(h1 @ W2 + b2)
    gemm_relu_wmma<HID, HID><<<256, 256, 0, stream>>>(
        h1, W2, b2, h2, ntiles);

    // 4) out = sigmoid(partial + h2 @ W3 + b3)
    deepfm_final<<<B_ROWS / 256, 256, 0, stream>>>(
        h2, W3, b3, partial, out);
}